// JaCDEManual_75024488726880
// MI455X (gfx1250) — compile-verified
//
#include <hip/hip_runtime.h>
#include <math.h>

typedef __attribute__((ext_vector_type(2))) float v2f;
typedef __attribute__((ext_vector_type(8))) float v8f;

#define NBLK 32
#define BDIM 512                  // 16 waves/block (wave32)
#define WAVES_PER_BLK (BDIM / 32)

// ---------------------------------------------------------------------------
// Accumulate a 16x16 f32 tile: OUT[b0:b0+16, i0:i0+16] += ACT · Wᵀ
// via V_WMMA_F32_16X16X4_F32 (wave32, full f32 precision, K-step = 4).
// A 16x4 layout: lane = 16*half + m ; VGPRs hold K = {2*half, 2*half+1}
// -> one float2 load per lane per k-step; B = Wᵀ symmetric.
// ---------------------------------------------------------------------------
__device__ __forceinline__ v8f wmma_tile(v8f acc,
                                         const float* __restrict__ act,
                                         const float* __restrict__ W,
                                         int K, int b0, int i0, int lane) {
  const int half = lane >> 4;
  const int l16  = lane & 15;
  const int k0   = half * 2;
  const float* ar = act + (size_t)(b0 + l16) * K + k0;
  const float* wr = W   + (size_t)(i0 + l16) * K + k0;
#pragma unroll 4
  for (int kk = 0; kk < K; kk += 4) {
    v2f a = *(const v2f*)(ar + kk);
    v2f b = *(const v2f*)(wr + kk);
    acc = __builtin_amdgcn_wmma_f32_16x16x4_f32(false, a, false, b,
                                                (short)0, acc, false, false);
  }
  return acc;
}

// ---------------------------------------------------------------------------
// Device-wide generation barrier (32 resident blocks; cannot deadlock).
// ---------------------------------------------------------------------------
__device__ __forceinline__ void grid_sync(unsigned int* cnt, unsigned int target) {
  __syncthreads();
  if (threadIdx.x == 0) {
    __threadfence();   // publish this block's phase writes device-wide
    __hip_atomic_fetch_add(cnt, 1u, __ATOMIC_RELEASE, __HIP_MEMORY_SCOPE_AGENT);
    while (__hip_atomic_load(cnt, __ATOMIC_ACQUIRE, __HIP_MEMORY_SCOPE_AGENT) < target)
      __builtin_amdgcn_s_sleep(1);
  }
  __syncthreads();
}

__global__ void zero_counter(unsigned int* cnt) { *cnt = 0u; }

// ---------------------------------------------------------------------------
// One persistent kernel for the whole Neumann-series ODE-RHS.
// Per-wave register state: drelu_r, dth_r (gate values of this wave's tile)
// and hd (h_dot accumulator) — the tile<->wave mapping is fixed across phases.
// ---------------------------------------------------------------------------
__global__ void __launch_bounds__(BDIM)
fused_kernel(const float* __restrict__ tp,      const float* __restrict__ h,
             const float* __restrict__ coeffs,  const float* __restrict__ dcoeffs,
             const float* __restrict__ tobs,    const float* __restrict__ wx,
             const float* __restrict__ wh,      const float* __restrict__ wout,
             const float* __restrict__ bias0,   const float* __restrict__ bias1,
             float* __restrict__ hdot,
             float* __restrict__ x,   float* __restrict__ xdot,
             float* __restrict__ relu, float* __restrict__ u, float* __restrict__ curr,
             unsigned int* cnt, int N, int B, int C, int H, int K_TERMS) {
  // ---- phase 0: cubic spline eval (one element per thread) ----
  {
    int id = blockIdx.x * BDIM + threadIdx.x;
    if (id < B * C) {
      int b = id / C, c = id % C;
      float t = tp[0];
      int cntLE = 0;
      for (int i = 0; i < N; ++i) cntLE += (tobs[i] <= t) ? 1 : 0;  // searchsorted 'right'
      int idx = cntLE - 1;
      if (idx < 0) idx = 0;
      if (idx > N - 2) idx = N - 2;
      float dt = t - tobs[idx];
      const float* cb = coeffs  + ((size_t)b * (N - 1) + idx) * 4 * C + c;
      const float* db = dcoeffs + ((size_t)b * (N - 1) + idx) * 4 * C + c;
      x[id]    = cb[0] + dt * (cb[(size_t)C] + dt * (cb[2 * (size_t)C] + dt * cb[3 * (size_t)C]));
      xdot[id] = db[0] + dt * (db[(size_t)C] + dt * (db[2 * (size_t)C] + dt * db[3 * (size_t)C]));
    }
  }
  unsigned int bar = 0;
  grid_sync(cnt, ++bar * NBLK);

  // Fixed tile coordinates for this wave (512 waves == 512 tiles).
  const int lane = threadIdx.x & 31;
  const int wid  = threadIdx.x >> 5;
  const int tid  = blockIdx.x * WAVES_PER_BLK + wid;
  const int tpr  = H >> 4;
  const int b0   = (tid / tpr) * 16;
  const int i0   = (tid % tpr) * 16;
  const int half = lane >> 4, l16 = lane & 15;
  const int col  = i0 + l16;

  v8f drelu_r = {}, dth_r = {}, hd = {};

  // ---- phase 1: l1 = x@wxT + h@whT + b0 -> relu (ws), drelu (registers) ----
  {
    v8f acc = {};
    acc = wmma_tile(acc, x, wx, C, b0, i0, lane);
    acc = wmma_tile(acc, h, wh, H, b0, i0, lane);
    const float bb = bias0[col];
#pragma unroll
    for (int v = 0; v < 8; ++v) {
      int row = b0 + half * 8 + v;
      float l1 = acc[v] + bb;
      relu[(size_t)row * H + col] = fmaxf(l1, 0.0f);
      drelu_r[v] = 1.0f / (1.0f + expf(-l1));
    }
  }
  grid_sync(cnt, ++bar * NBLK);      // relu is read cross-wave next

  // ---- phase 2: dtanh = 1 - tanh^2(relu@woutT + b1)  (registers only) ----
  {
    v8f acc = {};
    acc = wmma_tile(acc, relu, wout, H, b0, i0, lane);
    const float bb = bias1[col];
#pragma unroll
    for (int v = 0; v < 8; ++v) {
      float th = tanhf(acc[v] + bb);
      dth_r[v] = 1.0f - th * th;
    }
  }
  // no barrier: dth_r is private; next phase touches only xdot (ready) and u.

  // ---- phase 3: u0 = drelu ⊙ (xdot@wxT) ----
  {
    v8f acc = {};
    acc = wmma_tile(acc, xdot, wx, C, b0, i0, lane);
#pragma unroll
    for (int v = 0; v < 8; ++v) {
      int row = b0 + half * 8 + v;
      u[(size_t)row * H + col] = drelu_r[v] * acc[v];
    }
  }
  grid_sync(cnt, ++bar * NBLK);      // u read cross-wave next

  // ---- phase 4: curr0 = dtanh ⊙ (u@woutT) ; hd = curr0 ----
  {
    v8f acc = {};
    acc = wmma_tile(acc, u, wout, H, b0, i0, lane);
#pragma unroll
    for (int v = 0; v < 8; ++v) {
      int row = b0 + half * 8 + v;
      float c = dth_r[v] * acc[v];
      curr[(size_t)row * H + col] = c;
      hd[v] = c;
    }
  }
  grid_sync(cnt, ++bar * NBLK);

  // ---- Neumann terms: curr_k = dtanh ⊙ ((drelu ⊙ (curr@whT)) @ woutT) ----
  for (int k = 0; k < K_TERMS; ++k) {
    {
      v8f acc = {};
      acc = wmma_tile(acc, curr, wh, H, b0, i0, lane);
#pragma unroll
      for (int v = 0; v < 8; ++v) {
        int row = b0 + half * 8 + v;
        u[(size_t)row * H + col] = drelu_r[v] * acc[v];
      }
    }
    grid_sync(cnt, ++bar * NBLK);    // u read cross-wave next
    {
      v8f acc = {};
      acc = wmma_tile(acc, u, wout, H, b0, i0, lane);
#pragma unroll
      for (int v = 0; v < 8; ++v) {
        int row = b0 + half * 8 + v;
        float c = dth_r[v] * acc[v];
        hd[v] += c;
        curr[(size_t)row * H + col] = c;
      }
    }
    if (k < K_TERMS - 1)
      grid_sync(cnt, ++bar * NBLK);  // curr read cross-wave next iteration
  }

  // ---- final: write h_dot (each wave owns its tile; no sync needed) ----
#pragma unroll
  for (int v = 0; v < 8; ++v) {
    int row = b0 + half * 8 + v;
    hdot[(size_t)row * H + col] = hd[v];
  }
}

// ---------------------------------------------------------------------------
extern "C" void kernel_launch(void* const* d_in, const int* in_sizes, int n_in,
                              void* d_out, int out_size, void* d_ws, size_t ws_size,
                              hipStream_t stream) {
  const float* t       = (const float*)d_in[0];
  const float* h       = (const float*)d_in[1];
  const float* coeffs  = (const float*)d_in[2];
  const float* dcoeffs = (const float*)d_in[3];
  const float* tobs    = (const float*)d_in[4];
  const float* wx      = (const float*)d_in[5];
  const float* wh      = (const float*)d_in[6];
  const float* wout    = (const float*)d_in[7];
  const float* bias0   = (const float*)d_in[8];
  const float* bias1   = (const float*)d_in[9];
  float* hdot = (float*)d_out;

  const int N = in_sizes[4];        // 100
  const int H = in_sizes[8];        // 512
  const int C = in_sizes[5] / H;    // 64
  const int B = in_sizes[1] / H;    // 256
  const int K_TERMS = 8;

  float* ws    = (float*)d_ws;
  float* x     = ws; ws += (size_t)B * C;
  float* xdot  = ws; ws += (size_t)B * C;
  float* relu  = ws; ws += (size_t)B * H;
  float* u     = ws; ws += (size_t)B * H;
  float* curr  = ws; ws += (size_t)B * H;
  unsigned int* cnt = (unsigned int*)ws;

  zero_counter<<<1, 1, 0, stream>>>(cnt);
  fused_kernel<<<NBLK, BDIM, 0, stream>>>(
      t, h, coeffs, dcoeffs, tobs, wx, wh, wout, bias0, bias1,
      hdot, x, xdot, relu, u, curr, cnt, N, B, C, H, K_TERMS);
}